// WeightedKappaLoss_8186207666308
// MI455X (gfx1250) — compile-verified
//
#include <hip/hip_runtime.h>
#include <hip/hip_bf16.h>

typedef __attribute__((ext_vector_type(2))) float v2f;
typedef __attribute__((ext_vector_type(8))) float v8f;

#define NCLS 5
#define NBINS 25
#define WAVES_PER_BLOCK 8
#define HIST_BLOCKS 2048
#define HIST_THREADS 256

// ---------------------------------------------------------------------------
// soft-argmax -> rounded class index for one row of 5 logits
// matches: clip(round(softmax(p) @ [0..4]), 0, 4); rintf = round-half-even
// ---------------------------------------------------------------------------
__device__ __forceinline__ int pred_index(float a0, float a1, float a2,
                                          float a3, float a4) {
    float m  = fmaxf(fmaxf(fmaxf(a0, a1), fmaxf(a2, a3)), a4);
    float e0 = __expf(a0 - m);
    float e1 = __expf(a1 - m);
    float e2 = __expf(a2 - m);
    float e3 = __expf(a3 - m);
    float e4 = __expf(a4 - m);
    float s  = e0 + e1 + e2 + e3 + e4;
    float d  = e1 + 2.0f * e2 + 3.0f * e3 + 4.0f * e4;
    int   i  = (int)rintf(d / s);
    i = i < 0 ? 0 : i;
    i = i > (NCLS - 1) ? (NCLS - 1) : i;
    return i;
}

__global__ void kappa_zero_kernel(unsigned int* __restrict__ gconf) {
    if (threadIdx.x < NBINS) gconf[threadIdx.x] = 0u;
}

// ---------------------------------------------------------------------------
// Phase 1: streaming confusion-matrix histogram. HBM-bound (~192MB at 23.3TB/s).
// Each thread consumes 4 rows (= 80 contiguous bytes, 16B aligned) per chunk:
// five global_load_b128 for preds + one for labels. Per-wave LDS histogram
// copies (wave32 -> 8 copies/block, 32-int stride) reduce ds_add_u32 clashes.
// ---------------------------------------------------------------------------
__global__ void kappa_hist_kernel(const float* __restrict__ preds,
                                  const int* __restrict__ labels,
                                  unsigned int* __restrict__ gconf,
                                  int chunks, int n) {
    __shared__ unsigned int h[WAVES_PER_BLOCK * 32];

    const int tid = threadIdx.x;
    for (int i = tid; i < WAVES_PER_BLOCK * 32; i += blockDim.x) h[i] = 0u;
    __syncthreads();

    unsigned int* hw = &h[(tid >> 5) * 32];  // this wave's private copy

    const int gid    = blockIdx.x * blockDim.x + tid;
    const int stride = gridDim.x * blockDim.x;

    for (int c = gid; c < chunks; c += stride) {
        const float4* p = (const float4*)(preds + (size_t)c * 20);
        float4 p0 = p[0];
        float4 p1 = p[1];
        float4 p2 = p[2];
        float4 p3 = p[3];
        float4 p4 = p[4];
        int4 t = *(const int4*)(labels + (size_t)c * 4);

        int b0 = NCLS * t.x + pred_index(p0.x, p0.y, p0.z, p0.w, p1.x);
        int b1 = NCLS * t.y + pred_index(p1.y, p1.z, p1.w, p2.x, p2.y);
        int b2 = NCLS * t.z + pred_index(p2.z, p2.w, p3.x, p3.y, p3.z);
        int b3 = NCLS * t.w + pred_index(p3.w, p4.x, p4.y, p4.z, p4.w);

        atomicAdd(&hw[b0], 1u);
        atomicAdd(&hw[b1], 1u);
        atomicAdd(&hw[b2], 1u);
        atomicAdd(&hw[b3], 1u);
    }

    // Tail rows (n % 4) — empty for N = 8M but kept for generality.
    if (blockIdx.x == 0 && tid == 0) {
        for (int r = chunks * 4; r < n; ++r) {
            const float* q = preds + (size_t)r * NCLS;
            int b = NCLS * labels[r] + pred_index(q[0], q[1], q[2], q[3], q[4]);
            atomicAdd(&gconf[b], 1u);
        }
    }
    __syncthreads();

    // Fold the 8 per-wave copies, one global atomic per bin per block.
    for (int b = tid; b < NBINS; b += blockDim.x) {
        unsigned int s = 0u;
        #pragma unroll
        for (int w = 0; w < WAVES_PER_BLOCK; ++w) s += h[w * 32 + b];
        if (s) atomicAdd(&gconf[b], s);
    }
}

// ---------------------------------------------------------------------------
// Phase 2: single-wave finalize.
//   num = sum_ij conf[i][j] * w[i][j]
//   den = th^T * W * ph  computed with two V_WMMA_F32_16X16X4_F32 ops:
//         first covers K (=j) 0..3, second the j=4 column.
// A layout (ISA 16x4 f32): lane L -> M = L&15; VGPR0 holds K = 2*(L>>4),
// VGPR1 holds K+1. B: VGPR0 lanes0-15 = row K=0, lanes16-31 = row K=2;
// VGPR1 rows K=1/K=3. D: lane 0 holds column N=0, M = acc element index.
// result = N * num / den
// ---------------------------------------------------------------------------
__device__ __forceinline__ float kw(int i, int j) {
    float d = (float)(i - j);
    return d * d * (1.0f / 16.0f);  // (i-j)^2 / (C-1)^2, C=5
}

__global__ void kappa_finalize_kernel(const unsigned int* __restrict__ gconf,
                                      float* __restrict__ out, int n) {
    const int lane = threadIdx.x;  // 32 threads, one wave

    float c[NBINS];
    #pragma unroll
    for (int i = 0; i < NBINS; ++i) c[i] = (float)gconf[i];

    float th[NCLS], ph[NCLS];
    #pragma unroll
    for (int i = 0; i < NCLS; ++i) {
        float s = 0.0f;
        #pragma unroll
        for (int j = 0; j < NCLS; ++j) s += c[i * NCLS + j];
        th[i] = s;
    }
    #pragma unroll
    for (int j = 0; j < NCLS; ++j) {
        float s = 0.0f;
        #pragma unroll
        for (int i = 0; i < NCLS; ++i) s += c[i * NCLS + j];
        ph[j] = s;
    }

    float num = 0.0f;
    #pragma unroll
    for (int i = 0; i < NCLS; ++i)
        #pragma unroll
        for (int j = 0; j < NCLS; ++j) num += c[i * NCLS + j] * kw(i, j);

    // ---- den via WMMA: A[m][k] = th[m]*w[m][k], B[k][0] = ph[k] ----
    const int M  = lane & 15;
    const int K0 = (lane >> 4) * 2;

    v2f a1 = {0.0f, 0.0f}, b1 = {0.0f, 0.0f};
    if (M < NCLS) {
        a1.x = th[M] * kw(M, K0);
        a1.y = th[M] * kw(M, K0 + 1);
    }
    if ((lane & 15) == 0) {   // column N=0 of B
        int k = (lane < 16) ? 0 : 2;
        b1.x = ph[k];
        b1.y = ph[k + 1];
    }

    v8f acc = {};
    acc = __builtin_amdgcn_wmma_f32_16x16x4_f32(
        /*neg_a=*/false, a1, /*neg_b=*/false, b1,
        /*c_mod=*/(short)0, acc, /*reuse_a=*/false, /*reuse_b=*/false);

    // second K-slab: j = 4 only (K=0 slot)
    v2f a2 = {0.0f, 0.0f}, b2 = {0.0f, 0.0f};
    if (M < NCLS && K0 == 0) a2.x = th[M] * kw(M, 4);
    if (lane == 0)           b2.x = ph[4];

    acc = __builtin_amdgcn_wmma_f32_16x16x4_f32(
        false, a2, false, b2, (short)0, acc, false, false);

    if (lane == 0) {
        // D[m][0] for m=0..4 lives in acc[0..4] of lane 0
        float den = acc[0] + acc[1] + acc[2] + acc[3] + acc[4];
        out[0] = (float)n * num / den;
    }
}

// ---------------------------------------------------------------------------
extern "C" void kernel_launch(void* const* d_in, const int* in_sizes, int n_in,
                              void* d_out, int out_size, void* d_ws, size_t ws_size,
                              hipStream_t stream) {
    const float* preds  = (const float*)d_in[0];
    const int*   labels = (const int*)d_in[1];
    float*       out    = (float*)d_out;
    unsigned int* gconf = (unsigned int*)d_ws;  // 25 uint32 counters

    const int n      = in_sizes[1];  // number of rows (labels count)
    const int chunks = n / 4;

    kappa_zero_kernel<<<1, 32, 0, stream>>>(gconf);
    kappa_hist_kernel<<<HIST_BLOCKS, HIST_THREADS, 0, stream>>>(
        preds, labels, gconf, chunks, n);
    kappa_finalize_kernel<<<1, 32, 0, stream>>>(gconf, out, n);
}